// MultiPositiveContrastiveLoss_10969346474122
// MI455X (gfx1250) — compile-verified
//
#include <hip/hip_runtime.h>
#include <math.h>

// Problem constants (match reference)
#define B_ANCH 2048
#define P_POS  4
#define DIM    256
#define NNEG   32768
#define INV_T  20.0f        // 1 / 0.05
#define ALPHA  0.1f
#define EPS_N  1e-12f
#define SHIFT  20.0f        // max possible logit |cos|/T

typedef float v2f __attribute__((ext_vector_type(2)));
typedef float v8f __attribute__((ext_vector_type(8)));

// ---------------- wave reduction helpers (wave32) ----------------
__device__ inline float waveReduceSum32(float v) {
    #pragma unroll
    for (int off = 16; off > 0; off >>= 1)
        v += __shfl_xor(v, off, 32);
    return v;
}

__device__ inline float logaddexpf_dev(float a, float b) {
    float mx = fmaxf(a, b);
    float mn = fminf(a, b);
    return mx + log1pf(__expf(mn - mx));
}

__device__ inline void wait_asynccnt0() {
#if __has_builtin(__builtin_amdgcn_s_wait_asynccnt)
    __builtin_amdgcn_s_wait_asynccnt(0);
#else
    asm volatile("s_wait_asynccnt 0x0" ::: "memory");
#endif
}

// per-lane async 16B copy: global -> LDS (gfx1250 GLOBAL_LOAD_ASYNC_TO_LDS_B128)
__device__ inline void async_g2l_b128(unsigned lds_byte_addr, const void* gaddr) {
    asm volatile("global_load_async_to_lds_b128 %0, %1, off"
                 :: "v"(lds_byte_addr),
                    "v"((unsigned long long)(uintptr_t)gaddr)
                 : "memory");
}

// ---------------- kernel 1: row norms ----------------
__global__ __launch_bounds__(256)
void norm_kernel(const float* __restrict__ anch,
                 const float* __restrict__ neg,
                 float* __restrict__ a_norm,
                 float* __restrict__ inv_n) {
    int wave = (blockIdx.x * blockDim.x + threadIdx.x) >> 5;
    int lane = threadIdx.x & 31;
    if (wave < B_ANCH) {
        const float4* src = (const float4*)(anch + (size_t)wave * DIM) + lane * 2;
        float4 x0 = src[0], x1 = src[1];
        float ss = x0.x*x0.x + x0.y*x0.y + x0.z*x0.z + x0.w*x0.w
                 + x1.x*x1.x + x1.y*x1.y + x1.z*x1.z + x1.w*x1.w;
        ss = waveReduceSum32(ss);
        float inv = 1.0f / fmaxf(sqrtf(ss), EPS_N);
        float4* dst = (float4*)(a_norm + (size_t)wave * DIM) + lane * 2;
        x0.x *= inv; x0.y *= inv; x0.z *= inv; x0.w *= inv;
        x1.x *= inv; x1.y *= inv; x1.z *= inv; x1.w *= inv;
        dst[0] = x0; dst[1] = x1;
    } else if (wave < B_ANCH + NNEG) {
        int r = wave - B_ANCH;
        const float4* src = (const float4*)(neg + (size_t)r * DIM) + lane * 2;
        float4 x0 = src[0], x1 = src[1];
        float ss = x0.x*x0.x + x0.y*x0.y + x0.z*x0.z + x0.w*x0.w
                 + x1.x*x1.x + x1.y*x1.y + x1.z*x1.z + x1.w*x1.w;
        ss = waveReduceSum32(ss);
        if (lane == 0)
            inv_n[r] = 1.0f / fmaxf(sqrtf(ss), EPS_N);
    }
}

// ---------------- kernel 2: positive similarities ----------------
__global__ __launch_bounds__(256)
void possim_kernel(const float* __restrict__ a_norm,
                   const float* __restrict__ pos,
                   float* __restrict__ pos_sim) {
    int wv   = (blockIdx.x * blockDim.x + threadIdx.x) >> 5;
    int lane = threadIdx.x & 31;
    if (wv >= B_ANCH * P_POS) return;
    int b = wv >> 2;
    const float4* ps = (const float4*)(pos    + (size_t)wv * DIM) + lane * 2;
    const float4* as = (const float4*)(a_norm + (size_t)b  * DIM) + lane * 2;
    float4 p0 = ps[0], p1 = ps[1];
    float4 a0 = as[0], a1 = as[1];
    float ss  = p0.x*p0.x + p0.y*p0.y + p0.z*p0.z + p0.w*p0.w
              + p1.x*p1.x + p1.y*p1.y + p1.z*p1.z + p1.w*p1.w;
    float dot = a0.x*p0.x + a0.y*p0.y + a0.z*p0.z + a0.w*p0.w
              + a1.x*p1.x + a1.y*p1.y + a1.z*p1.z + a1.w*p1.w;
    ss  = waveReduceSum32(ss);
    dot = waveReduceSum32(dot);
    if (lane == 0) {
        float inv = 1.0f / fmaxf(sqrtf(ss), EPS_N);
        pos_sim[wv] = dot * inv * INV_T;
    }
}

// ---------------- kernel 3: fused fp32-WMMA GEMM + sum(exp) ----------------
#define WAVES    8
#define MTILE    32           // two 16-row A tiles per WG
#define NCHUNKS  16
#define ITERS    8            // NCHUNKS*ITERS*WAVES*32 == NNEG
#define LDSSTR   260          // pad: 16B-aligned rows; 16 row-lanes -> 16 banks (4 apart)

__global__ __launch_bounds__(256)
void gemm_lse_kernel(const float* __restrict__ a_norm,
                     const float* __restrict__ neg,
                     const float* __restrict__ inv_n,
                     float* __restrict__ sumexp) {
    __shared__ float Atile[MTILE * LDSSTR];
    __shared__ float rowsum[MTILE];

    const int tid = threadIdx.x;
    const int m0  = blockIdx.y * MTILE;

    // stage normalized 32x256 A tile into LDS via async global->LDS b128 copies
    {
        int r  = tid >> 3;            // 0..31 row (8 threads per row)
        int cf = (tid & 7) * 32;      // 32 consecutive floats per thread
        const float* gp = a_norm + (size_t)(m0 + r) * DIM + cf;
        unsigned lp = (unsigned)(uintptr_t)(Atile + r * LDSSTR + cf);
        #pragma unroll
        for (int j = 0; j < 8; ++j)
            async_g2l_b128(lp + j * 16u, gp + j * 4);
    }
    if (tid < MTILE) rowsum[tid] = 0.0f;
    wait_asynccnt0();          // wave waits for its own async transfers
    __syncthreads();           // tile visible WG-wide

    const int wave = tid >> 5;
    const int lane = tid & 31;
    const int rrow = lane & 15;           // row/col within 16-wide group
    const int hk   = (lane >> 4) << 1;    // 0 (K0,K1) lanes 0-15; 2 (K2,K3) lanes 16-31
    const float* Arow0 = Atile + rrow * LDSSTR;          // rows 0..15
    const float* Arow1 = Atile + (16 + rrow) * LDSSTR;   // rows 16..31

    // per-lane running exp-sums for both row tiles
    float esum0[8], esum1[8];
    #pragma unroll
    for (int i = 0; i < 8; ++i) { esum0[i] = 0.0f; esum1[i] = 0.0f; }

    for (int it = 0; it < ITERS; ++it) {
        // each wave: two adjacent 16-column tiles (32 columns)
        int tile32 = (blockIdx.x * ITERS + it) * WAVES + wave;   // 0 .. NNEG/32-1
        int c0   = tile32 * 32;
        int col0 = c0 + rrow;
        int col1 = c0 + 16 + rrow;
        float sn0 = inv_n[col0];
        float sn1 = inv_n[col1];
        const float* B0 = neg + (size_t)col0 * DIM;
        const float* B1 = neg + (size_t)col1 * DIM;

        v8f acc00 = {}, acc01 = {}, acc10 = {}, acc11 = {};

        #pragma unroll 1
        for (int k0 = 0; k0 < DIM; k0 += 16) {
            // distinct buffers -> independent loads -> clause + overlapped waits
            float2 q0[4], q1[4];
            v2f    a0[4], a1[4];
            #pragma unroll
            for (int j = 0; j < 4; ++j) q0[j] = *(const float2*)(B0 + k0 + hk + 4*j);
            #pragma unroll
            for (int j = 0; j < 4; ++j) q1[j] = *(const float2*)(B1 + k0 + hk + 4*j);
            #pragma unroll
            for (int j = 0; j < 4; ++j) a0[j] = *(const v2f*)(Arow0 + k0 + hk + 4*j);
            #pragma unroll
            for (int j = 0; j < 4; ++j) a1[j] = *(const v2f*)(Arow1 + k0 + hk + 4*j);
            #pragma unroll
            for (int j = 0; j < 4; ++j) {
                v2f b0; b0.x = q0[j].x * sn0; b0.y = q0[j].y * sn0;
                v2f b1; b1.x = q1[j].x * sn1; b1.y = q1[j].y * sn1;
                acc00 = __builtin_amdgcn_wmma_f32_16x16x4_f32(
                          false, a0[j], false, b0, (short)0, acc00, false, false);
                acc01 = __builtin_amdgcn_wmma_f32_16x16x4_f32(
                          false, a0[j], false, b1, (short)0, acc01, false, false);
                acc10 = __builtin_amdgcn_wmma_f32_16x16x4_f32(
                          false, a1[j], false, b0, (short)0, acc10, false, false);
                acc11 = __builtin_amdgcn_wmma_f32_16x16x4_f32(
                          false, a1[j], false, b1, (short)0, acc11, false, false);
            }
        }

        // fixed-shift exp, accumulate in registers (no cross-lane work per tile)
        #pragma unroll
        for (int i = 0; i < 8; ++i) {
            esum0[i] += __expf(acc00[i] * INV_T - SHIFT)
                      + __expf(acc01[i] * INV_T - SHIFT);
            esum1[i] += __expf(acc10[i] * INV_T - SHIFT)
                      + __expf(acc11[i] * INV_T - SHIFT);
        }
    }

    // single cross-lane reduction per workgroup
    // C layout: element i -> row i (lanes 0-15) / row i+8 (lanes 16-31)
    #pragma unroll
    for (int i = 0; i < 8; ++i) {
        float e0 = esum0[i];
        float e1 = esum1[i];
        #pragma unroll
        for (int off = 8; off > 0; off >>= 1) {
            e0 += __shfl_xor(e0, off, 16);
            e1 += __shfl_xor(e1, off, 16);
        }
        if (rrow == 0) {
            int rbase = i + ((lane >> 4) << 3);
            atomicAdd(&rowsum[rbase],      e0);   // ds_add_f32
            atomicAdd(&rowsum[16 + rbase], e1);
        }
    }
    __syncthreads();
    if (tid < MTILE)
        atomicAdd(&sumexp[m0 + tid], rowsum[tid]);   // global_atomic_add_f32
}

// ---------------- kernel 4: finalize loss (single block) ----------------
__global__ __launch_bounds__(256)
void finalize_kernel(const float* __restrict__ pos_sim,
                     const float* __restrict__ sumexp,
                     const int* __restrict__ counts,
                     float* __restrict__ out) {
    __shared__ float red[256];
    int t = threadIdx.x;
    float tot = 0.0f;
    for (int b = t; b < B_ANCH; b += 256) {
        float lse = SHIFT + logf(sumexp[b]);
        int   cnt = counts[b];
        float ps[P_POS];
        #pragma unroll
        for (int p = 0; p < P_POS; ++p) ps[p] = pos_sim[b * P_POS + p];
        #pragma unroll
        for (int p = 0; p < P_POS; ++p)
            if (p < cnt) tot += logaddexpf_dev(ps[p], lse) - ps[p];
        if (cnt > 1) {
            float m = fmaxf(fmaxf(ps[0], ps[1]), fmaxf(ps[2], ps[3]));
            float wsum = 0.0f, wdot = 0.0f;
            #pragma unroll
            for (int p = 0; p < P_POS; ++p) {
                float e = __expf(ps[p] - m);
                wsum += e; wdot += e * ps[p];
            }
            float wps = wdot / wsum;
            tot += ALPHA * (logaddexpf_dev(wps, lse) - wps);
        }
    }
    red[t] = tot;
    __syncthreads();
    for (int s = 128; s > 0; s >>= 1) {
        if (t < s) red[t] += red[t + s];
        __syncthreads();
    }
    if (t == 0) out[0] = red[0] / (float)B_ANCH;
}

// ---------------- host launcher ----------------
extern "C" void kernel_launch(void* const* d_in, const int* in_sizes, int n_in,
                              void* d_out, int out_size, void* d_ws, size_t ws_size,
                              hipStream_t stream) {
    const float* anch = (const float*)d_in[0];
    const float* pos  = (const float*)d_in[1];
    const float* neg  = (const float*)d_in[2];
    const int*   cnts = (const int*)d_in[3];

    float* ws      = (float*)d_ws;
    float* a_norm  = ws;                             // B*D
    float* inv_n   = a_norm  + (size_t)B_ANCH * DIM; // NNEG
    float* pos_sim = inv_n   + NNEG;                 // B*P
    float* sumexp  = pos_sim + B_ANCH * P_POS;       // B

    norm_kernel<<<(B_ANCH + NNEG) / 8, 256, 0, stream>>>(anch, neg, a_norm, inv_n);
    possim_kernel<<<(B_ANCH * P_POS) / 8, 256, 0, stream>>>(a_norm, pos, pos_sim);
    hipMemsetAsync(sumexp, 0, B_ANCH * sizeof(float), stream);
    gemm_lse_kernel<<<dim3(NCHUNKS, B_ANCH / MTILE), 256, 0, stream>>>(a_norm, neg, inv_n, sumexp);
    finalize_kernel<<<1, 256, 0, stream>>>(pos_sim, sumexp, cnts, (float*)d_out);
}